// MultiLossFactory_1597727834375
// MI455X (gfx1250) — compile-verified
//
#include <hip/hip_runtime.h>
#include <math.h>

#define RED_BLOCKS 2048
#define RED_THREADS 256

typedef __attribute__((ext_vector_type(2))) float v2f;
typedef __attribute__((ext_vector_type(4))) float v4f;
typedef __attribute__((ext_vector_type(8))) float v8f;

// Full wave32 sum via V_WMMA_F32_16X16X4_F32.
// A (16x4 f32, 2 VGPRs/lane): a.x = acc (K=0 for lanes 0-15, K=2 for lanes 16-31),
// a.y = 0. B = all ones => D[m][n] = acc[m] + acc[m+16] for every column n.
// Lane <16 holds rows 0..7 in its 8 D VGPRs, lane >=16 holds rows 8..15;
// one xor-16 shuffle completes the full 32-lane sum (exact f32 adds).
__device__ __forceinline__ float wave_sum32(float acc) {
    v2f a; a.x = acc; a.y = 0.0f;
    v2f b; b.x = 1.0f; b.y = 1.0f;
    v8f c = {};
    c = __builtin_amdgcn_wmma_f32_16x16x4_f32(
        /*neg_a=*/false, a, /*neg_b=*/false, b,
        /*c_mod=*/(short)0, c, /*reuse_a=*/false, /*reuse_b=*/false);
    float r = c[0] + c[1] + c[2] + c[3] + c[4] + c[5] + c[6] + c[7];
    r += __shfl_xor(r, 16, 32);
    return r;
}

// ---------------- Kernel 1: streaming masked-MSE partial sums ----------------
// Pure HBM-bandwidth bound (~214 MB single-use reads): non-temporal b128 loads
// keep the 192MB L2 from being churned by data with zero reuse.
__global__ void heat_mse_partial(const float* __restrict__ p,
                                 const float* __restrict__ g,
                                 const float* __restrict__ m,
                                 float* __restrict__ partial,
                                 long long n4) {
    __shared__ float lds[RED_THREADS / 32];
    float acc = 0.0f;
    const long long stride = (long long)gridDim.x * blockDim.x;
    const v4f* p4 = (const v4f*)p;
    const v4f* g4 = (const v4f*)g;
    const v4f* m4 = (const v4f*)m;
    for (long long i = (long long)blockIdx.x * blockDim.x + threadIdx.x; i < n4; i += stride) {
        v4f pv = __builtin_nontemporal_load(p4 + i);
        v4f gv = __builtin_nontemporal_load(g4 + i);
        v4f mv = __builtin_nontemporal_load(m4 + i);
        float dx = pv[0] - gv[0], dy = pv[1] - gv[1];
        float dz = pv[2] - gv[2], dw = pv[3] - gv[3];
        acc = fmaf(dx * dx, mv[0], acc);
        acc = fmaf(dy * dy, mv[1], acc);
        acc = fmaf(dz * dz, mv[2], acc);
        acc = fmaf(dw * dw, mv[3], acc);
    }
    // reconverged here: EXEC all ones for WMMA
    float ws = wave_sum32(acc);
    const int wid = threadIdx.x >> 5;
    if ((threadIdx.x & 31) == 0) lds[wid] = ws;
    __syncthreads();
    if (threadIdx.x == 0) {
        float s = 0.0f;
        for (int i = 0; i < RED_THREADS / 32; ++i) s += lds[i];
        partial[blockIdx.x] = s;
    }
}

// ---------------- Kernel 2: AE push/pull, one wave32 per image ----------------
__global__ void ae_loss_kernel(const float* __restrict__ tags,   // [B, N]
                               const int* __restrict__ joints,   // [B, M, K, 2]
                               float* __restrict__ out_push,     // [B]
                               float* __restrict__ out_pull,     // [B]
                               int Mn, int Kn, long long Nn) {
    const int b = blockIdx.x;
    const int t = threadIdx.x;  // 0..31, person id
    const float* tag = tags + (long long)b * Nn;
    const int* jb = joints + (long long)b * Mn * Kn * 2;

    float cnt = 0.0f, sum = 0.0f;
    if (t < Mn) {
        for (int k = 0; k < Kn; ++k) {
            int idx = jb[(t * Kn + k) * 2 + 0];
            int vis = jb[(t * Kn + k) * 2 + 1];
            float v = (vis > 0) ? 1.0f : 0.0f;
            sum = fmaf(tag[idx], v, sum);
            cnt += v;
        }
    }
    const float valid = (cnt > 0.0f) ? 1.0f : 0.0f;
    const float safe_cnt = fmaxf(cnt, 1.0f);
    const float mu = sum / safe_cnt;

    float pull_pp = 0.0f;
    if (t < Mn) {
        for (int k = 0; k < Kn; ++k) {
            int idx = jb[(t * Kn + k) * 2 + 0];
            int vis = jb[(t * Kn + k) * 2 + 1];
            float v = (vis > 0) ? 1.0f : 0.0f;
            float d = tag[idx] - mu;
            pull_pp = fmaf(d * d, v, pull_pp);
        }
        pull_pp /= safe_cnt;
    }

    // n = number of valid persons (wave sum, fixed xor-tree: deterministic)
    float n = valid;
    for (int o = 16; o >= 1; o >>= 1) n += __shfl_xor(n, o, 32);

    // pull = sum(valid ? pull_pp : 0) / max(n,1)
    float pl = valid * pull_pp;
    for (int o = 16; o >= 1; o >>= 1) pl += __shfl_xor(pl, o, 32);
    const float pull = pl / fmaxf(n, 1.0f);

    // push: s = sum over valid (i,j) of exp(-(mu_i-mu_j)^2), incl. diagonal
    float rowsum = 0.0f;
    for (int j = 0; j < Mn; ++j) {
        float muj = __shfl(mu, j, 32);
        float vj  = __shfl(valid, j, 32);
        float d = mu - muj;
        rowsum = fmaf(vj, __expf(-(d * d)), rowsum);
    }
    float s = valid * rowsum;
    for (int o = 16; o >= 1; o >>= 1) s += __shfl_xor(s, o, 32);
    const float denom = fmaxf((n - 1.0f) * n, 1.0f);
    const float push = (n >= 2.0f) ? (s - n) / denom * 0.5f : 0.0f;

    if (t == 0) {
        out_push[b] = push;
        out_pull[b] = pull;
    }
}

// ---------------- Kernel 3: deterministic finalize ----------------
__global__ void finalize_kernel(const float* __restrict__ partial, int nPartial,
                                const float* __restrict__ pushv,
                                const float* __restrict__ pullv, int Bn,
                                const float* __restrict__ p, const float* __restrict__ g,
                                const float* __restrict__ m,
                                long long nTail0, long long nTotal,
                                float* __restrict__ out) {
    __shared__ float lds[RED_THREADS / 32];
    float acc = 0.0f;
    for (int i = threadIdx.x; i < nPartial; i += blockDim.x) acc += partial[i];
    float ws = wave_sum32(acc);
    const int wid = threadIdx.x >> 5;
    if ((threadIdx.x & 31) == 0) lds[wid] = ws;
    __syncthreads();
    if (threadIdx.x == 0) {
        float s = 0.0f;
        for (int i = 0; i < RED_THREADS / 32; ++i) s += lds[i];
        // scalar tail if element count not divisible by 4 (0 iters for these shapes)
        for (long long i = nTail0; i < nTotal; ++i) {
            float d = p[i] - g[i];
            s = fmaf(d * d, m[i], s);
        }
        out[0] = (s / (float)nTotal) * 1.0f;  // HEATMAPS_LOSS_FACTOR
        float ps = 0.0f, pls = 0.0f;
        for (int i = 0; i < Bn; ++i) { ps += pushv[i]; pls += pullv[i]; }
        out[1] = (ps / (float)Bn) * 0.001f;   // PUSH_LOSS_FACTOR
        out[2] = (pls / (float)Bn) * 0.001f;  // PULL_LOSS_FACTOR
    }
}

extern "C" void kernel_launch(void* const* d_in, const int* in_sizes, int n_in,
                              void* d_out, int out_size, void* d_ws, size_t ws_size,
                              hipStream_t stream) {
    const float* hp   = (const float*)d_in[0];  // heatmaps_pred [B,K,H,W]
    const float* hg   = (const float*)d_in[1];  // heatmaps
    const float* hm   = (const float*)d_in[2];  // masks
    const float* tags = (const float*)d_in[3];  // tags_pred [B,K,H,W]
    const int* joints = (const int*)d_in[4];    // [B,M,K,2]

    const long long n = (long long)in_sizes[0];     // B*K*H*W
    const long long n4 = n / 4;
    const int Mn = 30, Kn = 17;
    const int Bn = in_sizes[4] / (Mn * Kn * 2);     // 16
    const long long Nn = n / Bn;                     // K*H*W

    float* partial = (float*)d_ws;                   // [RED_BLOCKS]
    float* pushb   = partial + RED_BLOCKS;           // [Bn]
    float* pullb   = pushb + Bn;                     // [Bn]

    heat_mse_partial<<<RED_BLOCKS, RED_THREADS, 0, stream>>>(hp, hg, hm, partial, n4);
    ae_loss_kernel<<<Bn, 32, 0, stream>>>(tags, joints, pushb, pullb, Mn, Kn, Nn);
    finalize_kernel<<<1, RED_THREADS, 0, stream>>>(partial, RED_BLOCKS, pushb, pullb, Bn,
                                                   hp, hg, hm, n4 * 4, n, (float*)d_out);
}